// GRUNet_3315714753140
// MI455X (gfx1250) — compile-verified
//
#include <hip/hip_runtime.h>

// Problem constants (match reference)
#define BATCH 2048
#define TT    512
#define INF   16
#define HH    20
#define GG    60
#define NL    5
#define FCH   64
#define OUTD  3
#define TILE  16   // batch rows per workgroup

typedef float v2f __attribute__((ext_vector_type(2)));
typedef float v8f __attribute__((ext_vector_type(8)));

// fp32 WMMA: D(16x16,f32) = A(16x4,f32) * B(4x16,f32) + C
// A/B: 2 floats per lane; C/D: 8 floats per lane (wave32).
__device__ __forceinline__ v8f wmma4(v2f a, v2f b, v8f c) {
  return __builtin_amdgcn_wmma_f32_16x16x4_f32(
      /*neg_a=*/false, a, /*neg_b=*/false, b,
      /*c_mod=*/(short)0, c, /*reuse_a=*/false, /*reuse_b=*/false);
}

__device__ __forceinline__ float sigmoidf_(float v) {
  return 1.0f / (1.0f + __expf(-v));
}

// ---------------------------------------------------------------------------
// Fused 5-layer GRU, layer-pipelined across 5 waves per workgroup.
// Workgroup = 16 batch rows. Wave l executes layer l at t = s - l.
// LDS: biases + double-buffered inter-layer activations + gate scratch.
// Weight B-fragments are held in registers for the entire time loop.
// ---------------------------------------------------------------------------
__global__ __launch_bounds__(NL * 32)
void gru_fused_kernel(const float* __restrict__ x,
                      const float* __restrict__ w_ih0, const float* __restrict__ w_hh0,
                      const float* __restrict__ b_ih0, const float* __restrict__ b_hh0,
                      const float* __restrict__ w_ih_r, const float* __restrict__ w_hh_r,
                      const float* __restrict__ b_ih_r, const float* __restrict__ b_hh_r,
                      float* __restrict__ hT)
{
  // float offsets into shared
  //  bih:  [5][64]            @ 0     (320)
  //  bhh:  [5][64]            @ 320   (320)
  //  ybuf: [5][2][16][20]     @ 640   (3200)  inter-layer / recurrent state
  //  gxs:  [5][64][16]        @ 3840  (5120)  x-path gate scratch (col-major)
  //  ghs:  [5][64][16]        @ 8960  (5120)  h-path gate scratch (col-major)
  __shared__ float sh[14080];
  const int OFF_BIH = 0, OFF_BHH = 320, OFF_Y = 640, OFF_GX = 3840, OFF_GH = 8960;

  const int tid  = threadIdx.x;
  const int wave = tid >> 5;
  const int lane = tid & 31;
  const int m    = lane & 15;   // batch row within tile (A/B/C column-lane id)
  const int kh   = lane >> 4;   // half-select (K/row split across lane halves)
  const int b0   = blockIdx.x * TILE;

  // cooperative init: biases into LDS, zero the activation ring (h_{-1} = 0)
  for (int i = tid; i < NL * GG; i += NL * 32) {
    int l = i / GG, n = i % GG;
    sh[OFF_BIH + l * 64 + n] = (l == 0) ? b_ih0[n] : b_ih_r[(l - 1) * GG + n];
    sh[OFF_BHH + l * 64 + n] = (l == 0) ? b_hh0[n] : b_hh_r[(l - 1) * GG + n];
  }
  for (int i = tid; i < NL * 2 * TILE * HH; i += NL * 32) sh[OFF_Y + i] = 0.0f;
  __syncthreads();

  const int l   = wave;
  const int Kin = (l == 0) ? INF : HH;
  const int nCx = Kin >> 2;            // K-chunks for the input path (4 or 5)

  // ---- preload loop-invariant weight B-fragments into registers ----
  // B(4x16) lane layout: n = nt*16 + (lane&15); k = chunk*4 + 2*(lane>>4) + {0,1}
  v2f bxf[5][4], bhf[5][4];
  {
    const float* wx = (l == 0) ? w_ih0 : (w_ih_r + (size_t)(l - 1) * GG * HH);
    const int sx = (l == 0) ? INF : HH;
    for (int c = 0; c < 5; ++c) {
      int k = c * 4 + 2 * kh;
      for (int nt = 0; nt < 4; ++nt) {
        int n = nt * 16 + m;
        bool ok = (n < GG) && (c < nCx);
        bxf[c][nt][0] = ok ? wx[n * sx + k] : 0.0f;
        bxf[c][nt][1] = ok ? wx[n * sx + k + 1] : 0.0f;
      }
    }
    const float* wh = (l == 0) ? w_hh0 : (w_hh_r + (size_t)(l - 1) * GG * HH);
    for (int c = 0; c < 5; ++c) {
      int k = c * 4 + 2 * kh;
      for (int nt = 0; nt < 4; ++nt) {
        int n = nt * 16 + m;
        bhf[c][nt][0] = (n < GG) ? wh[n * HH + k] : 0.0f;
        bhf[c][nt][1] = (n < GG) ? wh[n * HH + k + 1] : 0.0f;
      }
    }
  }

  float* gxs = &sh[OFF_GX + l * 1024];
  float* ghs = &sh[OFF_GH + l * 1024];
  const float* bih = &sh[OFF_BIH + l * 64];
  const float* bhh = &sh[OFF_BHH + l * 64];

  // ---- pipelined time loop ----
  for (int s = 0; s < TT + NL - 1; ++s) {
    const int t = s - l;                 // wave-uniform
    if (t >= 0 && t < TT) {
      v8f gx[4] = {};                    // x-path gates, 4 N-tiles (cols 0..63)
      v8f gh[4] = {};                    // h-path gates

      // input path: layer 0 reads x from global; others read layer l-1's LDS tile
      if (l == 0) {
        const float* px = x + ((size_t)(b0 + m) * TT + t) * INF + 2 * kh;
        if (t + 1 < TT) __builtin_prefetch(px + INF, 0, 1);  // next timestep
        #pragma unroll
        for (int c = 0; c < 4; ++c) {
          v2f a; a[0] = px[c * 4]; a[1] = px[c * 4 + 1];
          #pragma unroll
          for (int nt = 0; nt < 4; ++nt) gx[nt] = wmma4(a, bxf[c][nt], gx[nt]);
        }
      } else {
        const float* py = &sh[OFF_Y + (((l - 1) * 2 + (t & 1)) * TILE + m) * HH];
        #pragma unroll
        for (int c = 0; c < 5; ++c) {
          int k = c * 4 + 2 * kh;
          v2f a; a[0] = py[k]; a[1] = py[k + 1];
          #pragma unroll
          for (int nt = 0; nt < 4; ++nt) gx[nt] = wmma4(a, bxf[c][nt], gx[nt]);
        }
      }

      // hidden path: h_{t-1} lives in this layer's other buffer (zeroed at t=0)
      {
        const float* ph = &sh[OFF_Y + ((l * 2 + ((t + 1) & 1)) * TILE + m) * HH];
        #pragma unroll
        for (int c = 0; c < 5; ++c) {
          int k = c * 4 + 2 * kh;
          v2f a; a[0] = ph[k]; a[1] = ph[k + 1];
          #pragma unroll
          for (int nt = 0; nt < 4; ++nt) gh[nt] = wmma4(a, bhf[c][nt], gh[nt]);
        }
      }

      // dump C fragments column-major [col][row]: rows 8*kh..8*kh+7 are
      // contiguous -> two b128 stores per tile per lane
      #pragma unroll
      for (int nt = 0; nt < 4; ++nt) {
        float* dx = &gxs[(nt * 16 + m) * 16 + 8 * kh];
        ((float4*)dx)[0] = ((const float4*)&gx[nt])[0];
        ((float4*)dx)[1] = ((const float4*)&gx[nt])[1];
        float* dh = &ghs[(nt * 16 + m) * 16 + 8 * kh];
        ((float4*)dh)[0] = ((const float4*)&gh[nt])[0];
        ((float4*)dh)[1] = ((const float4*)&gh[nt])[1];
      }

      // GRU gate math (PyTorch convention; b_hh_n stays inside r*(...))
      const float* hprev = &sh[OFF_Y + ((l * 2 + ((t + 1) & 1)) * TILE) * HH];
      float*       hnew  = &sh[OFF_Y + ((l * 2 + (t & 1)) * TILE) * HH];
      #pragma unroll
      for (int e = lane; e < TILE * HH; e += 32) {
        int row = e & 15, j = e >> 4;    // j = hidden index 0..19
        float r = sigmoidf_(gxs[j * 16 + row] + bih[j] +
                            ghs[j * 16 + row] + bhh[j]);
        float z = sigmoidf_(gxs[(HH + j) * 16 + row] + bih[HH + j] +
                            ghs[(HH + j) * 16 + row] + bhh[HH + j]);
        float n = tanhf(gxs[(2 * HH + j) * 16 + row] + bih[2 * HH + j] +
                        r * (ghs[(2 * HH + j) * 16 + row] + bhh[2 * HH + j]));
        float h = (1.0f - z) * n + z * hprev[row * HH + j];
        hnew[row * HH + j] = h;
        if (l == NL - 1 && t == TT - 1)
          hT[(size_t)(b0 + row) * HH + j] = h;   // only global output
      }
    }
    __syncthreads();   // publish ybuf[l][t&1] to wave l+1 for step s+1
  }
}

// ---------------------------------------------------------------------------
// MLP head on h_T: [16x20]->64 -> 3x(64->64) -> 3, LeakyReLU between.
// One wave per 16 batch rows; fp32 WMMA; activations ping-pong through LDS.
// ---------------------------------------------------------------------------
__global__ __launch_bounds__(32)
void head_kernel(const float* __restrict__ hT,
                 const float* __restrict__ hw0, const float* __restrict__ hb0,
                 const float* __restrict__ hw_mid, const float* __restrict__ hb_mid,
                 const float* __restrict__ hw_out, const float* __restrict__ hb_out,
                 float* __restrict__ out)
{
  __shared__ float act[2][FCH * TILE];   // [buf][col*16 + row]
  const int lane = threadIdx.x & 31;
  const int m  = lane & 15;
  const int kh = lane >> 4;
  const int b0 = blockIdx.x * TILE;

  // layer 0: [16x20] @ W0^T[20x64]
  v8f c[4] = {};
  {
    v2f bf[5][4];
    for (int cc = 0; cc < 5; ++cc) {
      int k = cc * 4 + 2 * kh;
      for (int nt = 0; nt < 4; ++nt) {
        int n = nt * 16 + m;
        bf[cc][nt][0] = hw0[n * HH + k];
        bf[cc][nt][1] = hw0[n * HH + k + 1];
      }
    }
    const float* ph = hT + (size_t)(b0 + m) * HH;
    #pragma unroll
    for (int cc = 0; cc < 5; ++cc) {
      int k = cc * 4 + 2 * kh;
      v2f a; a[0] = ph[k]; a[1] = ph[k + 1];
      #pragma unroll
      for (int nt = 0; nt < 4; ++nt) c[nt] = wmma4(a, bf[cc][nt], c[nt]);
    }
  }
  // bias + LeakyReLU in C-layout (per-element), stage to act[0]
  #pragma unroll
  for (int nt = 0; nt < 4; ++nt) {
    float bias = hb0[nt * 16 + m];
    #pragma unroll
    for (int r = 0; r < 8; ++r) {
      float v = c[nt][r] + bias;
      c[nt][r] = (v >= 0.0f) ? v : 0.01f * v;
    }
    float* d = &act[0][(nt * 16 + m) * 16 + 8 * kh];
    ((float4*)d)[0] = ((const float4*)&c[nt])[0];
    ((float4*)d)[1] = ((const float4*)&c[nt])[1];
  }

  // 3 mid layers: [16x64] @ Wm^T[64x64]
  for (int i = 0; i < 3; ++i) {
    const float* wm  = hw_mid + (size_t)i * FCH * FCH;
    const float* src = act[i & 1];
    float*       dst = act[(i & 1) ^ 1];
    v8f d4[4] = {};
    for (int cc = 0; cc < 16; ++cc) {
      int k = cc * 4 + 2 * kh;
      v2f a; a[0] = src[k * 16 + m]; a[1] = src[(k + 1) * 16 + m];
      #pragma unroll
      for (int nt = 0; nt < 4; ++nt) {
        int n = nt * 16 + m;
        v2f b; b[0] = wm[n * FCH + k]; b[1] = wm[n * FCH + k + 1];
        d4[nt] = wmma4(a, b, d4[nt]);
      }
    }
    #pragma unroll
    for (int nt = 0; nt < 4; ++nt) {
      float bias = hb_mid[i * FCH + nt * 16 + m];
      #pragma unroll
      for (int r = 0; r < 8; ++r) {
        float v = d4[nt][r] + bias;
        d4[nt][r] = (v >= 0.0f) ? v : 0.01f * v;
      }
      float* dd = &dst[(nt * 16 + m) * 16 + 8 * kh];
      ((float4*)dd)[0] = ((const float4*)&d4[nt])[0];
      ((float4*)dd)[1] = ((const float4*)&d4[nt])[1];
    }
  }

  // final [16x64] @ [64x3]: too narrow for WMMA, do 48 scalar dots
  const float* fin = act[1];   // after i=2, dst == act[1]
  for (int e = lane; e < TILE * OUTD; e += 32) {
    int row = e & 15, o = e >> 4;
    float sum = hb_out[o];
    #pragma unroll 8
    for (int k = 0; k < FCH; ++k) sum += hw_out[o * FCH + k] * fin[k * 16 + row];
    out[(size_t)(b0 + row) * OUTD + o] = sum;
  }
}

extern "C" void kernel_launch(void* const* d_in, const int* in_sizes, int n_in,
                              void* d_out, int out_size, void* d_ws, size_t ws_size,
                              hipStream_t stream) {
  const float* x      = (const float*)d_in[0];
  const float* w_ih0  = (const float*)d_in[1];
  const float* w_hh0  = (const float*)d_in[2];
  const float* b_ih0  = (const float*)d_in[3];
  const float* b_hh0  = (const float*)d_in[4];
  const float* w_ih_r = (const float*)d_in[5];
  const float* w_hh_r = (const float*)d_in[6];
  const float* b_ih_r = (const float*)d_in[7];
  const float* b_hh_r = (const float*)d_in[8];
  const float* hw0    = (const float*)d_in[9];
  const float* hb0    = (const float*)d_in[10];
  const float* hw_mid = (const float*)d_in[11];
  const float* hb_mid = (const float*)d_in[12];
  const float* hw_out = (const float*)d_in[13];
  const float* hb_out = (const float*)d_in[14];

  float* hT = (float*)d_ws;   // [BATCH][HH] = 160 KB scratch

  gru_fused_kernel<<<BATCH / TILE, NL * 32, 0, stream>>>(
      x, w_ih0, w_hh0, b_ih0, b_hh0, w_ih_r, w_hh_r, b_ih_r, b_hh_r, hT);

  head_kernel<<<BATCH / TILE, 32, 0, stream>>>(
      hT, hw0, hb0, hw_mid, hb_mid, hw_out, hb_out, (float*)d_out);
}